// MAHNLayer_78752520339986
// MI455X (gfx1250) — compile-verified
//
#include <hip/hip_runtime.h>

#define D_IN  128
#define D_OUT 32

typedef float v2f __attribute__((ext_vector_type(2)));
typedef float v8f __attribute__((ext_vector_type(8)));

// ---------------------------------------------------------------------------
// Kernel 1: h = relu(input @ W) using V_WMMA_F32_16X16X4_F32 (one wave per
// 16x16 output tile). W (128x32 f32 = 16 KB) staged in LDS per workgroup.
//
// Fragment layouts (ISA 7.12.2, wave32):
//  A (16x4 f32, 2 VGPRs): lanes 0-15 -> M=lane, {K=k, k+1}; lanes 16-31 ->
//      M=lane-16, {K=k+2, k+3}
//  B (4x16 f32, 2 VGPRs): lanes hold N columns, same K split by lane half
//  C/D (16x16 f32, 8 VGPRs): VGPR v, lanes 0-15 -> M=v, N=lane;
//      lanes 16-31 -> M=v+8, N=lane-16
// ---------------------------------------------------------------------------
__global__ __launch_bounds__(256) void gemm_relu_wmma(
    const float* __restrict__ in, const float* __restrict__ W,
    float* __restrict__ h, int n_nodes, int total_tiles)
{
    __shared__ float sW[D_IN * D_OUT];  // 16 KB of the 320 KB WGP LDS
    for (int i = threadIdx.x; i < D_IN * D_OUT; i += blockDim.x)
        sW[i] = W[i];
    __syncthreads();

    const int lane  = threadIdx.x & 31;
    const int wave  = threadIdx.x >> 5;
    const int tile  = blockIdx.x * (blockDim.x >> 5) + wave;
    if (tile >= total_tiles) return;

    const int ntilesN = D_OUT / 16;              // 2
    const int mtile = tile / ntilesN;
    const int ntile = tile - mtile * ntilesN;
    const int m0 = mtile * 16;
    const int n0 = ntile * 16;

    const int half = lane >> 4;                  // K-high selector
    const int l    = lane & 15;                  // M row (A) / N col (B)

    // clamp A row for safety when n_nodes % 16 != 0 (stores are guarded)
    int arow = m0 + l;
    if (arow >= n_nodes) arow = n_nodes - 1;
    const float* aptr = in + (size_t)arow * D_IN;

    v8f c = {};
    for (int k = 0; k < D_IN; k += 4) {
        v2f a;
        a.x = aptr[k + 2 * half + 0];
        a.y = aptr[k + 2 * half + 1];
        v2f b;
        b.x = sW[(k + 2 * half + 0) * D_OUT + n0 + l];
        b.y = sW[(k + 2 * half + 1) * D_OUT + n0 + l];
        // 8 args: (neg_a, A, neg_b, B, c_mod, C, reuse_a, reuse_b)
        c = __builtin_amdgcn_wmma_f32_16x16x4_f32(
                false, a, false, b, (short)0, c, false, false);
    }

    const int mbase = m0 + half * 8;
    #pragma unroll
    for (int v = 0; v < 8; ++v) {
        int m = mbase + v;
        if (m < n_nodes) {
            float x = c[v];
            h[(size_t)m * D_OUT + n0 + l] = x > 0.0f ? x : 0.0f;
        }
    }
}

// ---------------------------------------------------------------------------
// Kernel 0: zero the accumulation buffer (harness poisons d_out with 0xAA).
// ---------------------------------------------------------------------------
__global__ void zero_f32(float* __restrict__ p, int n)
{
    int i = blockIdx.x * blockDim.x + threadIdx.x;
    if (i < n) p[i] = 0.0f;
}

// ---------------------------------------------------------------------------
// Kernel 2: per-edge gather-scale-scatter. One wave32 per edge; lane == the
// feature index (D_OUT == 32 == warpSize). Edge scalars are wave-uniform
// loads; per-lane atomicAdd lowers to non-returning global_atomic_add_f32.
// h (12.8 MB) and out (12.8 MB) are L2-resident on MI455X (192 MB L2), so the
// ~410 MB of gather+atomic traffic stays on-chip; only the 19.2 MB edge
// stream hits HBM.
// ---------------------------------------------------------------------------
__global__ __launch_bounds__(256) void edge_scatter(
    const float* __restrict__ h, const float* __restrict__ dw1,
    const int* __restrict__ erow, const int* __restrict__ ecol,
    const int* __restrict__ etime, float* __restrict__ out, int nedges)
{
    int gid  = blockIdx.x * blockDim.x + threadIdx.x;
    int e    = gid >> 5;
    int lane = gid & 31;
    if (e >= nedges) return;

    float decay = dw1[etime[e]];
    int   col   = ecol[e];
    int   row   = erow[e];

    float v = decay * h[(size_t)col * D_OUT + lane];
    atomicAdd(out + (size_t)row * D_OUT + lane, v);
}

// ---------------------------------------------------------------------------
// Kernel 3: per-node window decay. observation_time read on-device (1-elem
// array; no host copies permitted under graph capture).
// ---------------------------------------------------------------------------
__global__ void scale_out(
    float* __restrict__ out, const float* __restrict__ dw2,
    const int* __restrict__ arrive, const int* __restrict__ obs, int n_nodes)
{
    int gid  = blockIdx.x * blockDim.x + threadIdx.x;
    int node = gid >> 5;
    int lane = gid & 31;
    if (node >= n_nodes) return;

    int widx = 60 * obs[0] - arrive[node] - 1;
    out[(size_t)node * D_OUT + lane] *= dw2[widx];
}

extern "C" void kernel_launch(void* const* d_in, const int* in_sizes, int n_in,
                              void* d_out, int out_size, void* d_ws, size_t ws_size,
                              hipStream_t stream)
{
    const float* input  = (const float*)d_in[0];
    const float* W      = (const float*)d_in[1];
    const float* dw1    = (const float*)d_in[2];
    const float* dw2    = (const float*)d_in[3];
    const int*   erow   = (const int*)d_in[4];
    const int*   ecol   = (const int*)d_in[5];
    const int*   etime  = (const int*)d_in[6];
    const int*   arrive = (const int*)d_in[7];
    const int*   obs    = (const int*)d_in[8];

    const int N = in_sizes[0] / D_IN;
    const int E = in_sizes[4];

    float* out = (float*)d_out;
    float* h   = (float*)d_ws;      // N * D_OUT floats (12.8 MB scratch)

    // 0) zero the output accumulator
    {
        int n = N * D_OUT;
        zero_f32<<<(n + 255) / 256, 256, 0, stream>>>(out, n);
    }

    // 1) h = relu(input @ W)   (FP32 WMMA)
    {
        int total_tiles = ((N + 15) / 16) * (D_OUT / 16);
        int wavesPerBlk = 256 / 32;
        int blocks = (total_tiles + wavesPerBlk - 1) / wavesPerBlk;
        gemm_relu_wmma<<<blocks, 256, 0, stream>>>(input, W, h, N, total_tiles);
    }

    // 2) scatter-add edge messages
    {
        long long threads = (long long)E * 32;
        int blocks = (int)((threads + 255) / 256);
        edge_scatter<<<blocks, 256, 0, stream>>>(h, dw1, erow, ecol, etime, out, E);
    }

    // 3) per-node window decay scaling
    {
        long long threads = (long long)N * 32;
        int blocks = (int)((threads + 255) / 256);
        scale_out<<<blocks, 256, 0, stream>>>(out, dw2, arrive, obs, N);
    }
}